// ScientificLIFNeuron_14422500180574
// MI455X (gfx1250) — compile-verified
//
#include <hip/hip_runtime.h>

// LIF neuron update: pure streaming, HBM-bandwidth bound.
// 32 B/neuron mandatory traffic -> ~46 us floor @ 23.3 TB/s. Goal: saturate HBM.
// Strategy: b128 non-temporal loads/stores, 32-bit voffset + saddr (scale_offset)
// addressing, wave32 blocks, no LDS / no barriers.

typedef float v4f __attribute__((ext_vector_type(4)));

#define K_LEAK 0.1f      // DT / TAU
#define V_REST_C 0.0f
#define V_RESET_C 0.0f
#define REF_PERIOD_C 2.0f
#define ADAPT_UP 0.05f
#define ADAPT_DN 0.025f  // ADAPT_RATE * 0.5
#define TH_LO 0.4f       // V_TH * 0.8
#define TH_HI 0.75f      // V_TH * 1.5

__device__ __forceinline__ void lif_step(float cur, float v, float th, float lst,
                                         float tf,
                                         float& o_spike, float& o_v, float& o_th, float& o_lst) {
    const bool refractory = (tf - lst) < REF_PERIOD_C;
    // leaky integration (matches reference ordering: v + (rest - v)*k + I*k)
    const float v_int = v + (V_REST_C - v) * K_LEAK + cur * K_LEAK;
    const float v1 = refractory ? v : v_int;
    const bool sp = v1 >= th;
    o_spike = sp ? 1.0f : 0.0f;
    o_lst   = sp ? tf : lst;
    o_v     = sp ? V_RESET_C : v1;
    float tn = sp ? (th + ADAPT_UP) : (th - ADAPT_DN);
    tn = fminf(fmaxf(tn, TH_LO), TH_HI);
    o_th = tn;
}

// One float4 (4 neurons) per lane. 32-bit index -> GVS saddr+voffset addressing.
__global__ __launch_bounds__(256) void lif_vec4_kernel(
    const float* __restrict__ in_cur,
    const float* __restrict__ in_v,
    const float* __restrict__ in_th,
    const float* __restrict__ in_lst,
    const int*   __restrict__ t_ptr,
    float* __restrict__ out,   // [4, N] flattened
    unsigned nvec)             // N / 4  (fits in 32 bits: 8.4M)
{
    const unsigned i = blockIdx.x * 256u + threadIdx.x;
    if (i >= nvec) return;

    const float tf = (float)(*t_ptr);  // uniform -> scalar load, broadcast

    const v4f cur = __builtin_nontemporal_load((const v4f*)in_cur + i);
    const v4f v   = __builtin_nontemporal_load((const v4f*)in_v   + i);
    const v4f th  = __builtin_nontemporal_load((const v4f*)in_th  + i);
    const v4f lst = __builtin_nontemporal_load((const v4f*)in_lst + i);

    v4f o_spike, o_v, o_th, o_lst;
#pragma unroll
    for (int j = 0; j < 4; ++j) {
        float s, vn, tn, ln;
        lif_step(cur[j], v[j], th[j], lst[j], tf, s, vn, tn, ln);
        o_spike[j] = s; o_v[j] = vn; o_th[j] = tn; o_lst[j] = ln;
    }

    // Uniform row bases (SALU); per-lane part stays the same 32-bit scaled offset.
    v4f* o0 = (v4f*)out;            // spike
    v4f* o1 = o0 + nvec;            // v_new
    v4f* o2 = o1 + nvec;            // th_new
    v4f* o3 = o2 + nvec;            // lst_new
    __builtin_nontemporal_store(o_spike, o0 + i);
    __builtin_nontemporal_store(o_v,     o1 + i);
    __builtin_nontemporal_store(o_th,    o2 + i);
    __builtin_nontemporal_store(o_lst,   o3 + i);
}

// Scalar tail kernel for N % 4 != 0 (not hit for N = 2^25, kept for robustness).
__global__ void lif_tail_kernel(
    const float* __restrict__ in_cur,
    const float* __restrict__ in_v,
    const float* __restrict__ in_th,
    const float* __restrict__ in_lst,
    const int*   __restrict__ t_ptr,
    float* __restrict__ out,
    long long start, long long nelem)
{
    const long long i = start + (long long)blockIdx.x * blockDim.x + threadIdx.x;
    if (i >= nelem) return;
    const float tf = (float)(*t_ptr);
    float s, vn, tn, ln;
    lif_step(in_cur[i], in_v[i], in_th[i], in_lst[i], tf, s, vn, tn, ln);
    out[0 * nelem + i] = s;
    out[1 * nelem + i] = vn;
    out[2 * nelem + i] = tn;
    out[3 * nelem + i] = ln;
}

extern "C" void kernel_launch(void* const* d_in, const int* in_sizes, int n_in,
                              void* d_out, int out_size, void* d_ws, size_t ws_size,
                              hipStream_t stream) {
    const float* in_cur = (const float*)d_in[0];
    const float* in_v   = (const float*)d_in[1];
    const float* in_th  = (const float*)d_in[2];
    const float* in_lst = (const float*)d_in[3];
    const int*   t_ptr  = (const int*)d_in[4];
    float* out = (float*)d_out;

    const long long n    = (long long)in_sizes[0];
    const long long nvec = n / 4;
    const long long rem  = n - nvec * 4;

    if (nvec > 0) {
        const int block = 256;
        const long long grid = (nvec + block - 1) / block;
        lif_vec4_kernel<<<(unsigned)grid, block, 0, stream>>>(
            in_cur, in_v, in_th, in_lst, t_ptr, out, (unsigned)nvec);
    }
    if (rem > 0) {
        lif_tail_kernel<<<1, (unsigned)rem, 0, stream>>>(
            in_cur, in_v, in_th, in_lst, t_ptr, out, nvec * 4, n);
    }
}